// AttentionUnroll_352187318651
// MI455X (gfx1250) — compile-verified
//
#include <hip/hip_runtime.h>

namespace {
constexpr int H      = 32;
constexpr int D      = 64;
constexpr int S      = 4096;
constexpr int START  = 4080;
constexpr int NSTEP  = 16;
constexpr float ATT_SCALE = 0.125f;
constexpr int TROWS   = 64;          // K rows per async-staged tile
constexpr int NT      = S / TROWS;   // 64 tiles
constexpr int KSTRIDE = 68;          // padded LDS row stride (floats): bank-conflict-free
}

typedef __attribute__((ext_vector_type(2))) float v2f;
typedef __attribute__((ext_vector_type(4))) float v4f;
typedef __attribute__((ext_vector_type(8))) float v8f;
typedef int v4i_vs __attribute__((vector_size(16)));   // matches builtin prototype

#if __has_builtin(__builtin_amdgcn_global_load_async_to_lds_b128)
#define USE_ASYNC_K 1
#else
#define USE_ASYNC_K 0
#endif

#if USE_ASYNC_K
__device__ __forceinline__ void wait_async_le(void) {
#if __has_builtin(__builtin_amdgcn_s_wait_asynccnt)
  __builtin_amdgcn_s_wait_asynccnt(4);   // allow next tile in flight, retire older one
#else
  asm volatile("s_wait_asynccnt 4" ::: "memory");
#endif
}
__device__ __forceinline__ void wait_async_all(void) {
#if __has_builtin(__builtin_amdgcn_s_wait_asynccnt)
  __builtin_amdgcn_s_wait_asynccnt(0);
#else
  asm volatile("s_wait_asynccnt 0" ::: "memory");
#endif
}

// Stage one 64x64 f32 tile (16 KB) global -> LDS with row padding to KSTRIDE.
// 1024 b128 transfers, 4 per thread, each wave instruction fully coalesced (512B).
__device__ __forceinline__ void async_copy_tile(const float* __restrict__ gsrc,
                                                float* lbuf, int tid) {
#pragma unroll
  for (int j = 0; j < 4; ++j) {
    int e   = (j * 256 + tid) * 4;       // flat float index within tile
    int row = e >> 6;
    int col = e & 63;
    __attribute__((address_space(1))) v4i_vs* gp =
        (__attribute__((address_space(1))) v4i_vs*)(gsrc + e);
    __attribute__((address_space(3))) v4i_vs* lp =
        (__attribute__((address_space(3))) v4i_vs*)(lbuf + row * KSTRIDE + col);
    __builtin_amdgcn_global_load_async_to_lds_b128(gp, lp, 0, 0);
  }
}
#endif

__device__ __forceinline__ float blk_reduce_max(float val, float* red, int tid) {
  red[tid] = val;
  __syncthreads();
  for (int off = 128; off > 0; off >>= 1) {
    if (tid < off) red[tid] = fmaxf(red[tid], red[tid + off]);
    __syncthreads();
  }
  float r = red[0];
  __syncthreads();
  return r;
}

__device__ __forceinline__ float blk_reduce_sum(float val, float* red, int tid) {
  red[tid] = val;
  __syncthreads();
  for (int off = 128; off > 0; off >>= 1) {
    if (tid < off) red[tid] = red[tid] + red[tid + off];
    __syncthreads();
  }
  float r = red[0];
  __syncthreads();
  return r;
}

// out[e] = sum_d xin[d] * W[d*64 + e] for e in [e0, e0+16), via
// V_WMMA_F32_16X16X4_F32 with A = xin broadcast across the 16 M-rows.
// f32 A 16x4 layout: lanes 0-15 hold {k0,k1} in v[0],v[1]; lanes 16-31 hold {k2,k3}.
// f32 B 4x16 layout mirrors A's K split; column n = lane (mod 16).
__device__ __forceinline__ void proj64_wmma(const float* __restrict__ W,
                                            const float* xin, float* dst,
                                            int e0, int lane) {
  const int half = (lane >> 4) << 1;  // 0 for lanes 0-15, 2 for lanes 16-31
  const int n = lane & 15;
  v8f acc = {};
#pragma unroll
  for (int c = 0; c < 64; c += 4) {
    v2f a, b;
    a.x = xin[c + half];
    a.y = xin[c + half + 1];
    b.x = W[(c + half) * 64 + e0 + n];
    b.y = W[(c + half + 1) * 64 + e0 + n];
    // (neg_a, A, neg_b, B, c_mod, C, reuse_a, reuse_b)
    acc = __builtin_amdgcn_wmma_f32_16x16x4_f32(false, a, false, b, (short)0,
                                                acc, false, false);
  }
  // D rows are identical (A rows broadcast); lane n<16 holds out[e0+n] in acc[0].
  if (lane < 16) dst[e0 + n] = acc[0];
}

__global__ __launch_bounds__(256, 1)
void AttentionUnroll_352187318651_kernel(const float* __restrict__ x,
                                         const float* __restrict__ k,
                                         const float* __restrict__ v,
                                         const float* __restrict__ wq,
                                         const float* __restrict__ wk,
                                         const float* __restrict__ wv,
                                         const float* __restrict__ wo,
                                         float* __restrict__ out) {
  __shared__ __align__(16) float sScores[S];          // 16 KB
  __shared__ __align__(16) float sNewK[NSTEP * D];    // appended K rows
  __shared__ __align__(16) float sNewV[NSTEP * D];    // appended V rows
  __shared__ __align__(16) float sQ[D];
  __shared__ float sCtx[D];
  __shared__ float sX[D];
  __shared__ float sRed[256];
  __shared__ float sPart[256];
#if USE_ASYNC_K
  __shared__ __align__(16) float sBufK[2 * TROWS * KSTRIDE];  // ~34 KB double buffer
#endif

  const int tid  = threadIdx.x;
  const int wave = tid >> 5;
  const int lane = tid & 31;
  const int bh   = blockIdx.x;   // b*H + h
  const int h    = bh & (H - 1);

  const float* Kg = k + (size_t)bh * S * D;
  const float* Vg = v + (size_t)bh * S * D;

  if (tid < D) sX[tid] = x[(size_t)bh * D + tid];
  __syncthreads();

  for (int step = 0; step < NSTEP; ++step) {
    // ---- Phase 1: q / k_new / v_new projections (WMMA) ----
    for (int u = wave; u < 12; u += 8) {       // 3 projections x 4 e-tiles
      int pj = u >> 2;
      int e0 = (u & 3) << 4;
      const float* W = (pj == 0 ? wq : pj == 1 ? wk : wv) + h * (D * D);
      float* dst = (pj == 0 ? sQ : pj == 1 ? &sNewK[step * D] : &sNewV[step * D]);
      proj64_wmma(W, sX, dst, e0, lane);
    }
    __syncthreads();

    // ---- Phase 2: scores[s] = K[s,:] . q ----
#if USE_ASYNC_K
    {
      // Async-staged, double-buffered K tiles (ASYNCcnt pipeline).
      const int r  = tid >> 2;            // row within tile (0..63)
      const int qd = tid & 3;             // quarter of the row
      const int ch = qd << 4;             // starting column (0/16/32/48)
      async_copy_tile(Kg, &sBufK[0], tid);
      for (int t = 0; t < NT; ++t) {
        if (t + 1 < NT) {
          async_copy_tile(Kg + (size_t)(t + 1) * TROWS * D,
                          &sBufK[((t + 1) & 1) * (TROWS * KSTRIDE)], tid);
          wait_async_le();                // tile t fully landed in LDS
        } else {
          wait_async_all();
        }
        __syncthreads();
        int s  = t * TROWS + r;
        int rr = s - START;
        const float* kr = (rr >= 0 && rr <= step)
                              ? &sNewK[rr * D + ch]
                              : &sBufK[(t & 1) * (TROWS * KSTRIDE) + r * KSTRIDE + ch];
        float acc = 0.f;
#pragma unroll
        for (int j = 0; j < 4; ++j) {
          v4f kv = *(const v4f*)(kr + 4 * j);
          acc = fmaf(kv.x, sQ[ch + 4 * j + 0], acc);
          acc = fmaf(kv.y, sQ[ch + 4 * j + 1], acc);
          acc = fmaf(kv.z, sQ[ch + 4 * j + 2], acc);
          acc = fmaf(kv.w, sQ[ch + 4 * j + 3], acc);
        }
        acc += __shfl_xor(acc, 1);
        acc += __shfl_xor(acc, 2);
        if (qd == 0) sScores[s] = acc;
        __syncthreads();
      }
    }
#else
    {
      int wbase = wave << 9;                   // 512 rows per wave
      for (int c = 0; c < 16; ++c) {
        int s = wbase + (c << 5) + lane;
        float acc = 0.f;
        if (s + 32 < START) __builtin_prefetch(Kg + (size_t)(s + 32) * D, 0, 0);
        if (s < START) {
          const float* kr = Kg + (size_t)s * D;
#pragma unroll
          for (int j = 0; j < 16; ++j) {
            v4f kv = *(const v4f*)(kr + (j << 2));
            acc = fmaf(kv.x, sQ[4 * j + 0], acc);
            acc = fmaf(kv.y, sQ[4 * j + 1], acc);
            acc = fmaf(kv.z, sQ[4 * j + 2], acc);
            acc = fmaf(kv.w, sQ[4 * j + 3], acc);
          }
        } else {
          int r2 = s - START;
          const float* kr = (r2 <= step) ? (const float*)&sNewK[r2 * D]
                                         : (const float*)(Kg + (size_t)s * D);
#pragma unroll
          for (int j = 0; j < 16; ++j) {
            v4f kv = *(const v4f*)(kr + (j << 2));
            acc = fmaf(kv.x, sQ[4 * j + 0], acc);
            acc = fmaf(kv.y, sQ[4 * j + 1], acc);
            acc = fmaf(kv.z, sQ[4 * j + 2], acc);
            acc = fmaf(kv.w, sQ[4 * j + 3], acc);
          }
        }
        sScores[s] = acc;
      }
    }
#endif
    __syncthreads();

    // ---- Phase 3: softmax over S (scale folded in) ----
    float lmax = -3.402823466e38f;
    for (int s2 = tid; s2 < S; s2 += 256) lmax = fmaxf(lmax, sScores[s2]);
    float gmax = blk_reduce_max(lmax, sRed, tid);
    float lsum = 0.f;
    for (int s2 = tid; s2 < S; s2 += 256) {
      float p = __expf((sScores[s2] - gmax) * ATT_SCALE);
      sScores[s2] = p;
      lsum += p;
    }
    float gsum = blk_reduce_sum(lsum, sRed, tid);
    float inv = 1.0f / gsum;

    // ---- Phase 4: ctx[d] = (sum_s p[s] V[s,d]) * inv (coalesced row sweeps) ----
    {
      int dcol  = tid & 63;
      int g     = tid >> 6;
      int sbase = g << 10;
      int send  = sbase + 1024;
      float acc = 0.f;
      int smid = (send < START) ? send : START;
      for (int s2 = sbase; s2 < smid; ++s2) {
        if ((s2 & 31) == 0 && s2 + 32 < smid)
          __builtin_prefetch(Vg + (size_t)(s2 + 32) * D + dcol, 0, 0);
        acc = fmaf(sScores[s2], Vg[(size_t)s2 * D + dcol], acc);
      }
      for (int s2 = (sbase > START ? sbase : START); s2 < send; ++s2) {
        int r2 = s2 - START;
        float vv = (r2 <= step) ? sNewV[r2 * D + dcol] : Vg[(size_t)s2 * D + dcol];
        acc = fmaf(sScores[s2], vv, acc);
      }
      sPart[tid] = acc;
    }
    __syncthreads();
    if (tid < D) {
      sCtx[tid] = (sPart[tid] + sPart[64 + tid] + sPart[128 + tid] +
                   sPart[192 + tid]) * inv;
    }
    __syncthreads();

    // ---- Phase 5: x = ctx @ wo (WMMA) ----
    if (wave < 4) proj64_wmma(wo + h * (D * D), sCtx, sX, wave << 4, lane);
    __syncthreads();
  }

  if (tid < D) out[(size_t)bh * D + tid] = sX[tid];
}

extern "C" void kernel_launch(void* const* d_in, const int* in_sizes, int n_in,
                              void* d_out, int out_size, void* d_ws, size_t ws_size,
                              hipStream_t stream) {
  (void)in_sizes; (void)n_in; (void)d_ws; (void)ws_size; (void)out_size;
  const float* x  = (const float*)d_in[0];
  const float* k  = (const float*)d_in[1];
  const float* v  = (const float*)d_in[2];
  const float* wq = (const float*)d_in[3];
  const float* wk = (const float*)d_in[4];
  const float* wv = (const float*)d_in[5];
  const float* wo = (const float*)d_in[6];
  float* out = (float*)d_out;

  dim3 grid(32 * 32);  // one workgroup per (batch, head)
  dim3 block(256);     // 8 wave32
  hipLaunchKernelGGL(AttentionUnroll_352187318651_kernel, grid, block, 0, stream,
                     x, k, v, wq, wk, wv, wo, out);
}